// Transformer_43568148251442
// MI455X (gfx1250) — compile-verified
//
#include <hip/hip_runtime.h>
#include <hip/hip_bf16.h>

typedef __attribute__((ext_vector_type(2))) float v2f;
typedef __attribute__((ext_vector_type(8))) float v8f;

#define NB 8
#define NN 4096
#define ND 256
#define NDT 64
#define NK 16
#define PITCH 68   // 68 % 64 == 4 -> conflict-free when M varies across lanes

__device__ __forceinline__ v8f wmma4(v2f a, v2f b, v8f c) {
    // D = A(16x4 f32) * B(4x16 f32) + C(16x16 f32)
    return __builtin_amdgcn_wmma_f32_16x16x4_f32(false, a, false, b, (short)0, c,
                                                 false, false);
}

__device__ __forceinline__ v8f zero8() {
    v8f z = {0.f, 0.f, 0.f, 0.f, 0.f, 0.f, 0.f, 0.f};
    return z;
}

// ---------------------------------------------------------------- KNN
// grid = NB * (NN/256) = 128, block = 256, dyn LDS = 4*NN*4 = 64KB
__global__ void knn_kernel(const float* __restrict__ xyz, int* __restrict__ ids) {
    extern __shared__ float sm[];
    float* sx = sm;
    float* sy = sm + NN;
    float* sz = sm + 2 * NN;
    float* sq = sm + 3 * NN;
    int b = blockIdx.x >> 4;
    int n = ((blockIdx.x & 15) << 8) + threadIdx.x;
    for (int i = threadIdx.x; i < NN; i += 256) {
        float x = xyz[(b * NN + i) * 3 + 0];
        float y = xyz[(b * NN + i) * 3 + 1];
        float z = xyz[(b * NN + i) * 3 + 2];
        sx[i] = x; sy[i] = y; sz[i] = z;
        sq[i] = x * x + y * y + z * z;
    }
    __syncthreads();
    float bd[NK];
    int bi[NK];
#pragma unroll
    for (int j = 0; j < NK; ++j) { bd[j] = 3.4e38f; bi[j] = 0; }
    float px = sx[n], py = sy[n], pz = sz[n], ps = sq[n];
    for (int m = 0; m < NN; ++m) {
        float d = ps + sq[m] - 2.f * (px * sx[m] + py * sy[m] + pz * sz[m]);
        if (d < bd[NK - 1]) {
            int j = NK - 1;
            while (j > 0 && bd[j - 1] > d) {
                bd[j] = bd[j - 1]; bi[j] = bi[j - 1]; --j;
            }
            bd[j] = d; bi[j] = m;
        }
    }
#pragma unroll
    for (int j = 0; j < NK; ++j) ids[(b * NN + n) * NK + j] = bi[j];
}

// ---------------------------------------------------------------- map_in
// ft[b][n][dt] = sum_d fts[b][d][n] * Win[dt][d] + bin[dt]
// grid = NB * (NN/128) = 256, block = 256, dyn LDS = 16384*4 = 64KB (WinT)
__global__ void mapin_kernel(const float* __restrict__ fts, const float* __restrict__ Win,
                             const float* __restrict__ bin, float* __restrict__ ft) {
    extern __shared__ float sm[];
    float* sW = sm;  // [256][64] WinT
    int tid = threadIdx.x;
    int b = blockIdx.x >> 5;
    int m0blk = (blockIdx.x & 31) * 128;
    for (int l = tid; l < ND * NDT; l += 256) {
        int k = l >> 6, c = l & 63;
        sW[l] = Win[c * ND + k];
    }
    __syncthreads();
    int lane = tid & 31, w = tid >> 5;
    int LWv = lane & 15, LH = lane >> 4;
    int m0 = m0blk + w * 16;
    v8f acc[4];
#pragma unroll
    for (int ct = 0; ct < 4; ++ct) acc[ct] = zero8();
    for (int ks = 0; ks < 64; ++ks) {
        int kkb = ks * 4 + 2 * LH;
        v2f a;
        a.x = fts[(b * ND + kkb) * NN + m0 + LWv];
        a.y = fts[(b * ND + kkb + 1) * NN + m0 + LWv];
#pragma unroll
        for (int ct = 0; ct < 4; ++ct) {
            v2f bb;
            bb.x = sW[kkb * 64 + ct * 16 + LWv];
            bb.y = sW[(kkb + 1) * 64 + ct * 16 + LWv];
            acc[ct] = wmma4(a, bb, acc[ct]);
        }
    }
#pragma unroll
    for (int ct = 0; ct < 4; ++ct) {
        int dt = ct * 16 + LWv;
        float bias = bin[dt];
#pragma unroll
        for (int j = 0; j < 8; ++j) {
            int m = m0 + j + 8 * LH;
            ft[(b * NN + m) * NDT + dt] = acc[ct][j] + bias;
        }
    }
}

// ---------------------------------------------------------------- 64x64 linear
// C[r][n] = sum_k A[r][k]*W[n][k] + bias[n], rows = NB*NN contiguous
// grid = 256, block = 256, dyn LDS = (128*PITCH + 64*64)*4
__global__ void linear64_kernel(const float* __restrict__ A, const float* __restrict__ W,
                                const float* __restrict__ bias, float* __restrict__ C) {
    extern __shared__ float sm[];
    float* sA = sm;                  // [128][PITCH]
    float* sW = sm + 128 * PITCH;    // [64][64] WT
    int tid = threadIdx.x;
    int row0 = blockIdx.x * 128;
    for (int l = tid; l < 128 * 16; l += 256) {
        int r = l >> 4, i = l & 15;
        ((float4*)(sA + r * PITCH))[i] = ((const float4*)(A + (row0 + r) * 64))[i];
    }
    for (int l = tid; l < 4096; l += 256) {
        int k = l >> 6, c = l & 63;
        sW[l] = W[c * 64 + k];
    }
    __syncthreads();
    int lane = tid & 31, w = tid >> 5;
    int LWv = lane & 15, LH = lane >> 4;
    v8f acc[4];
#pragma unroll
    for (int ct = 0; ct < 4; ++ct) acc[ct] = zero8();
    for (int ks = 0; ks < 16; ++ks) {
        int kkb = ks * 4 + 2 * LH;
        v2f a;
        a.x = sA[(w * 16 + LWv) * PITCH + kkb];
        a.y = sA[(w * 16 + LWv) * PITCH + kkb + 1];
#pragma unroll
        for (int ct = 0; ct < 4; ++ct) {
            v2f bb;
            bb.x = sW[kkb * 64 + ct * 16 + LWv];
            bb.y = sW[(kkb + 1) * 64 + ct * 16 + LWv];
            acc[ct] = wmma4(a, bb, acc[ct]);
        }
    }
#pragma unroll
    for (int ct = 0; ct < 4; ++ct) {
        float bsv = bias[ct * 16 + LWv];
#pragma unroll
        for (int j = 0; j < 8; ++j) {
            int r = row0 + w * 16 + j + 8 * LH;
            C[r * 64 + ct * 16 + LWv] = acc[ct][j] + bsv;
        }
    }
}

// ---------------------------------------------------------------- fused attention
// one workgroup = 8 query points = 128 (n,k) rows; wave w owns rows [16w,16w+16)
// grid = NB*512 = 4096, block = 256
// LDS floats: sQ 512 | sP 8192 | sVL 8192 | sH 128*PITCH | sX 128*PITCH | sW 4096 | sId 128
__global__ void attn_kernel(const float* __restrict__ xyz, const int* __restrict__ ids,
                            const float* __restrict__ qf, const float* __restrict__ kf,
                            const float* __restrict__ vf,
                            const float* __restrict__ Wp1, const float* __restrict__ bp1,
                            const float* __restrict__ Wp2, const float* __restrict__ bp2,
                            const float* __restrict__ Wa1, const float* __restrict__ ba1,
                            const float* __restrict__ Wa2, const float* __restrict__ ba2,
                            float* __restrict__ attn_out, float* __restrict__ fo) {
    extern __shared__ float sm[];
    float* sQ  = sm;                       // [8][64]
    float* sP  = sQ + 512;                 // [128][64]
    float* sVL = sP + 8192;                // [128][64]
    float* sH  = sVL + 8192;               // [128][PITCH]
    float* sX  = sH + 128 * PITCH;         // [128][PITCH]  (hp1 / hidden / a)
    float* sW  = sX + 128 * PITCH;         // [64][64]
    int*   sId = (int*)(sW + 4096);        // [128]

    int tid = threadIdx.x;
    int b = blockIdx.x >> 9;
    int n0 = (blockIdx.x & 511) * 8;
    int lane = tid & 31, w = tid >> 5;
    int LWv = lane & 15, LH = lane >> 4;

    if (tid < 128) sId[tid] = ids[(b * NN + n0) * NK + tid];
    for (int l = tid; l < 512; l += 256) sQ[l] = qf[(b * NN + n0) * NDT + l];
    __syncthreads();

    // gather vl
    {
        int r = tid >> 1, c0 = (tid & 1) * 32;
        int id = sId[r];
        const float4* src = (const float4*)(vf + (b * NN + id) * NDT + c0);
        float4* dst = (float4*)(sVL + r * 64 + c0);
#pragma unroll
        for (int i = 0; i < 8; ++i) dst[i] = src[i];
    }
    // pos layer 1 (K=3, VALU): hp1 -> sX
    {
        int r = tid >> 1, c0 = (tid & 1) * 32;
        int id = sId[r];
        int nq = n0 + (r >> 4);
        float rx = xyz[(b * NN + id) * 3 + 0] - xyz[(b * NN + nq) * 3 + 0];
        float ry = xyz[(b * NN + id) * 3 + 1] - xyz[(b * NN + nq) * 3 + 1];
        float rz = xyz[(b * NN + id) * 3 + 2] - xyz[(b * NN + nq) * 3 + 2];
        for (int c = c0; c < c0 + 32; ++c) {
            float v = rx * Wp1[c * 3 + 0] + ry * Wp1[c * 3 + 1] + rz * Wp1[c * 3 + 2] + bp1[c];
            sX[r * PITCH + c] = fmaxf(v, 0.f);
        }
    }
    for (int l = tid; l < 4096; l += 256) sW[l] = Wp2[(l & 63) * 64 + (l >> 6)];
    __syncthreads();

    // pos = hp1 @ Wp2^T + bp2
    {
        v8f acc[4];
#pragma unroll
        for (int ct = 0; ct < 4; ++ct) acc[ct] = zero8();
        for (int ks = 0; ks < 16; ++ks) {
            int kkb = ks * 4 + 2 * LH;
            v2f a;
            a.x = sX[(w * 16 + LWv) * PITCH + kkb];
            a.y = sX[(w * 16 + LWv) * PITCH + kkb + 1];
#pragma unroll
            for (int ct = 0; ct < 4; ++ct) {
                v2f bb;
                bb.x = sW[kkb * 64 + ct * 16 + LWv];
                bb.y = sW[(kkb + 1) * 64 + ct * 16 + LWv];
                acc[ct] = wmma4(a, bb, acc[ct]);
            }
        }
#pragma unroll
        for (int ct = 0; ct < 4; ++ct) {
            float bsv = bp2[ct * 16 + LWv];
#pragma unroll
            for (int j = 0; j < 8; ++j)
                sP[(w * 16 + j + 8 * LH) * 64 + ct * 16 + LWv] = acc[ct][j] + bsv;
        }
    }
    __syncthreads();

    // h = q - kl + pos  -> sH
    {
        int r = tid >> 1, c0 = (tid & 1) * 32;
        int id = sId[r];
        int nl = r >> 4;
        const float* kr = kf + (b * NN + id) * NDT;
        for (int c = c0; c < c0 + 32; ++c)
            sH[r * PITCH + c] = sQ[nl * 64 + c] - kr[c] + sP[r * 64 + c];
    }
    __syncthreads();

    // attention MLP: 64 -> 256 (relu) -> 64, column-blocked by 64
    v8f accA[4];
#pragma unroll
    for (int ct = 0; ct < 4; ++ct) accA[ct] = zero8();
    for (int cb = 0; cb < 4; ++cb) {
        for (int l = tid; l < 4096; l += 256)
            sW[l] = Wa1[(cb * 64 + (l & 63)) * 64 + (l >> 6)];
        __syncthreads();
        // hidden = relu(h @ Wa1_cb^T + ba1_cb) -> sX
        {
            v8f acc[4];
#pragma unroll
            for (int ct = 0; ct < 4; ++ct) acc[ct] = zero8();
            for (int ks = 0; ks < 16; ++ks) {
                int kkb = ks * 4 + 2 * LH;
                v2f a;
                a.x = sH[(w * 16 + LWv) * PITCH + kkb];
                a.y = sH[(w * 16 + LWv) * PITCH + kkb + 1];
#pragma unroll
                for (int ct = 0; ct < 4; ++ct) {
                    v2f bb;
                    bb.x = sW[kkb * 64 + ct * 16 + LWv];
                    bb.y = sW[(kkb + 1) * 64 + ct * 16 + LWv];
                    acc[ct] = wmma4(a, bb, acc[ct]);
                }
            }
#pragma unroll
            for (int ct = 0; ct < 4; ++ct) {
                float bsv = ba1[cb * 64 + ct * 16 + LWv];
#pragma unroll
                for (int j = 0; j < 8; ++j) {
                    float v = acc[ct][j] + bsv;
                    sX[(w * 16 + j + 8 * LH) * PITCH + ct * 16 + LWv] = fmaxf(v, 0.f);
                }
            }
        }
        __syncthreads();
        for (int l = tid; l < 4096; l += 256)
            sW[l] = Wa2[(l & 63) * ND + cb * 64 + (l >> 6)];
        __syncthreads();
        // accA += hidden @ Wa2_cb
        for (int ks = 0; ks < 16; ++ks) {
            int kkb = ks * 4 + 2 * LH;
            v2f a;
            a.x = sX[(w * 16 + LWv) * PITCH + kkb];
            a.y = sX[(w * 16 + LWv) * PITCH + kkb + 1];
#pragma unroll
            for (int ct = 0; ct < 4; ++ct) {
                v2f bb;
                bb.x = sW[kkb * 64 + ct * 16 + LWv];
                bb.y = sW[(kkb + 1) * 64 + ct * 16 + LWv];
                accA[ct] = wmma4(a, bb, accA[ct]);
            }
        }
        __syncthreads();
    }

    // softmax over k: wave w's 16 rows == all k for query n0+w; M-dim = k
    float aout[4][8];
#pragma unroll
    for (int ct = 0; ct < 4; ++ct) {
        float bsv = ba2[ct * 16 + LWv];
        float t[8], mx = -3.4e38f;
#pragma unroll
        for (int j = 0; j < 8; ++j) {
            t[j] = (accA[ct][j] + bsv) * 0.125f;  // 1/sqrt(64)
            mx = fmaxf(mx, t[j]);
        }
        mx = fmaxf(mx, __shfl_xor(mx, 16, 32));
        float sum = 0.f;
#pragma unroll
        for (int j = 0; j < 8; ++j) {
            t[j] = __expf(t[j] - mx);
            sum += t[j];
        }
        sum += __shfl_xor(sum, 16, 32);
        float inv = 1.f / sum;
#pragma unroll
        for (int j = 0; j < 8; ++j) aout[ct][j] = t[j] * inv;
    }
    // a -> sX for transpose + weighted sum
#pragma unroll
    for (int ct = 0; ct < 4; ++ct)
#pragma unroll
        for (int j = 0; j < 8; ++j)
            sX[(w * 16 + j + 8 * LH) * PITCH + ct * 16 + LWv] = aout[ct][j];
    __syncthreads();

    // attn[b][dt][n][k] (k contiguous, 64B stores)
    for (int o = tid; o < 512; o += 256) {
        int nl = o >> 6, dt = o & 63;
        float t[16];
#pragma unroll
        for (int kk = 0; kk < 16; ++kk) t[kk] = sX[(nl * 16 + kk) * PITCH + dt];
        float4* d4 = (float4*)(attn_out + (((size_t)(b * NDT + dt)) * NN + n0 + nl) * NK);
#pragma unroll
        for (int i = 0; i < 4; ++i) {
            float4 v = {t[4 * i], t[4 * i + 1], t[4 * i + 2], t[4 * i + 3]};
            d4[i] = v;
        }
    }
    // fo[b][n][c] = sum_k a * (vl + pos)
    for (int o = tid; o < 512; o += 256) {
        int nl = o >> 6, c = o & 63;
        float s = 0.f;
#pragma unroll
        for (int kk = 0; kk < 16; ++kk) {
            int r = nl * 16 + kk;
            s += sX[r * PITCH + c] * (sVL[r * 64 + c] + sP[r * 64 + c]);
        }
        fo[(b * NN + n0 + nl) * NDT + c] = s;
    }
}

// ---------------------------------------------------------------- out projection
// out[b][d][n] = sum_c Wout[d][c]*fo[b][n][c] + bout[d] + fts[b][d][n]
// grid = NB*32 = 256, block = 256, LDS = (256*PITCH + 128*PITCH)*4
__global__ void outproj_kernel(const float* __restrict__ fo, const float* __restrict__ Wout,
                               const float* __restrict__ bout, const float* __restrict__ fts,
                               float* __restrict__ out) {
    extern __shared__ float sm[];
    float* sWo = sm;                   // [256][PITCH]
    float* sB  = sm + 256 * PITCH;     // [128][PITCH]
    int tid = threadIdx.x;
    int b = blockIdx.x >> 5;
    int n0blk = (blockIdx.x & 31) * 128;
    for (int l = tid; l < 256 * 16; l += 256) {
        int r = l >> 4, i = l & 15;
        ((float4*)(sWo + r * PITCH))[i] = ((const float4*)(Wout + r * 64))[i];
    }
    for (int l = tid; l < 128 * 16; l += 256) {
        int r = l >> 4, i = l & 15;
        ((float4*)(sB + r * PITCH))[i] = ((const float4*)(fo + (b * NN + n0blk + r) * 64))[i];
    }
    __syncthreads();
    int lane = tid & 31, w = tid >> 5;
    int LWv = lane & 15, LH = lane >> 4;
#pragma unroll
    for (int i = 0; i < 2; ++i) {
        int rt = w * 2 + i;  // d row-tile
        for (int ct = 0; ct < 8; ++ct) {
            v8f acc = zero8();
            for (int ks = 0; ks < 16; ++ks) {
                int kkb = ks * 4 + 2 * LH;
                v2f a;
                a.x = sWo[(rt * 16 + LWv) * PITCH + kkb];
                a.y = sWo[(rt * 16 + LWv) * PITCH + kkb + 1];
                v2f bb;
                bb.x = sB[(ct * 16 + LWv) * PITCH + kkb];
                bb.y = sB[(ct * 16 + LWv) * PITCH + kkb + 1];
                acc = wmma4(a, bb, acc);
            }
#pragma unroll
            for (int j = 0; j < 8; ++j) {
                int d = rt * 16 + j + 8 * LH;
                int n = n0blk + ct * 16 + LWv;
                size_t o = ((size_t)(b * ND + d)) * NN + n;
                out[o] = acc[j] + bout[d] + fts[o];
            }
        }
    }
}

// ---------------------------------------------------------------- launch
extern "C" void kernel_launch(void* const* d_in, const int* in_sizes, int n_in,
                              void* d_out, int out_size, void* d_ws, size_t ws_size,
                              hipStream_t stream) {
    (void)in_sizes; (void)n_in; (void)out_size; (void)ws_size;
    const float* xyz  = (const float*)d_in[0];
    const float* fts  = (const float*)d_in[1];
    const float* Win  = (const float*)d_in[3];
    const float* bin  = (const float*)d_in[4];
    const float* Wq   = (const float*)d_in[5];
    const float* bq   = (const float*)d_in[6];
    const float* Wk   = (const float*)d_in[7];
    const float* bk   = (const float*)d_in[8];
    const float* Wv   = (const float*)d_in[9];
    const float* bv   = (const float*)d_in[10];
    const float* Wp1  = (const float*)d_in[11];
    const float* bp1  = (const float*)d_in[12];
    const float* Wp2  = (const float*)d_in[13];
    const float* bp2  = (const float*)d_in[14];
    const float* Wa1  = (const float*)d_in[15];
    const float* ba1  = (const float*)d_in[16];
    const float* Wa2  = (const float*)d_in[17];
    const float* ba2  = (const float*)d_in[18];
    const float* Wout = (const float*)d_in[19];
    const float* bout = (const float*)d_in[20];

    float* outF = (float*)d_out;                 // fts_out: NB*ND*NN
    float* attn = outF + (size_t)NB * ND * NN;   // attn: NB*NDT*NN*NK

    int*   ids = (int*)d_ws;                       // NB*NN*NK = 524288 ints
    float* ftb = (float*)d_ws + 524288;            // NB*NN*NDT
    float* qf  = ftb + 2097152;
    float* kfb = qf + 2097152;
    float* vfb = kfb + 2097152;
    float* fo  = vfb + 2097152;

    knn_kernel<<<128, 256, 4 * NN * 4, stream>>>(xyz, ids);
    mapin_kernel<<<256, 256, ND * NDT * 4, stream>>>(fts, Win, bin, ftb);
    size_t lin_lds = (size_t)(128 * PITCH + 64 * 64) * 4;
    linear64_kernel<<<256, 256, lin_lds, stream>>>(ftb, Wq, bq, qf);
    linear64_kernel<<<256, 256, lin_lds, stream>>>(ftb, Wk, bk, kfb);
    linear64_kernel<<<256, 256, lin_lds, stream>>>(ftb, Wv, bv, vfb);
    size_t attn_lds = (size_t)(512 + 8192 + 8192 + 128 * PITCH + 128 * PITCH + 4096 + 128) * 4;
    attn_kernel<<<NB * 512, 256, attn_lds, stream>>>(xyz, ids, qf, kfb, vfb,
                                                     Wp1, bp1, Wp2, bp2,
                                                     Wa1, ba1, Wa2, ba2, attn, fo);
    size_t op_lds = (size_t)(256 * PITCH + 128 * PITCH) * 4;
    outproj_kernel<<<256, 256, op_lds, stream>>>(fo, Wout, bout, fts, outF);
}